// CustomCLIP_41455024341117
// MI455X (gfx1250) — compile-verified
//
#include <hip/hip_runtime.h>
#include <math.h>

typedef __attribute__((ext_vector_type(2))) float v2f;
typedef __attribute__((ext_vector_type(4))) float v4f;
typedef __attribute__((ext_vector_type(8))) float v8f;
typedef __attribute__((ext_vector_type(4))) int   v4i;

#define BN_EPS 1e-5f

// gfx1250 async global->LDS DMA (ASYNCcnt-tracked), guarded so the file
// always compiles; falls back to the VGPR round-trip path if absent.
// Probe-learned signature: (int4 AS1* gaddr, int4 AS3* laddr, imm offset, imm cpol).
#if __has_builtin(__builtin_amdgcn_global_load_async_to_lds_b128)
#define ASYNC_LDS 1
#else
#define ASYNC_LDS 0
#endif

typedef v4i __attribute__((address_space(1)))* g4ptr;
typedef v4i __attribute__((address_space(3)))* l4ptr;

__device__ __forceinline__ void async_copy_b128(const float* g, float* l) {
#if ASYNC_LDS
    __builtin_amdgcn_global_load_async_to_lds_b128((g4ptr)g, (l4ptr)l, 0, 0);
#else
    (void)g; (void)l;
#endif
}

__device__ __forceinline__ void async_wait0() {
#if ASYNC_LDS
#if __has_builtin(__builtin_amdgcn_s_wait_asynccnt)
    __builtin_amdgcn_s_wait_asynccnt(0);
#else
    asm volatile("s_wait_asynccnt 0" ::: "memory");
#endif
#endif
}

// Per-call scratch for BN statistics & top-k indices (re-initialized every
// call by k_stats_init / k_topk -> deterministic, no cross-call state).
__device__ float g_sum[768];
__device__ float g_sumsq[768];
__device__ float g_scale[768];
__device__ float g_shift[768];
__device__ int   g_topidx[256 * 6];

// ---------------------------------------------------------------- scalar exp
__global__ void k_scalar_exp(const float* __restrict__ ls, float* __restrict__ out) {
    if (threadIdx.x == 0) out[0] = expf(ls[0]);
}

// ------------------------------------------------------------- BN statistics
__global__ void k_stats_init(int C) {
    int c = blockIdx.x * blockDim.x + threadIdx.x;
    if (c < C) { g_sum[c] = 0.0f; g_sumsq[c] = 0.0f; }
}

__global__ void k_stats_accum(const float* __restrict__ A, int M, int C, int rowsPerBlock) {
    int r0 = blockIdx.x * rowsPerBlock;
    int r1 = r0 + rowsPerBlock; if (r1 > M) r1 = M;
    for (int c = threadIdx.x; c < C; c += blockDim.x) {
        float s = 0.0f, ss = 0.0f;
        for (int r = r0; r < r1; ++r) {
            float v = A[(size_t)r * C + c];
            s += v; ss += v * v;
        }
        atomicAdd(&g_sum[c], s);
        atomicAdd(&g_sumsq[c], ss);
    }
}

__global__ void k_stats_final(const float* __restrict__ gamma, const float* __restrict__ beta,
                              int C, float invM) {
    int c = blockIdx.x * blockDim.x + threadIdx.x;
    if (c < C) {
        float mean = g_sum[c] * invM;
        float var  = g_sumsq[c] * invM - mean * mean;   // biased var (matches jnp.var)
        float inv  = rsqrtf(var + BN_EPS);
        float sc   = gamma[c] * inv;
        g_scale[c] = sc;
        g_shift[c] = beta[c] - mean * sc;
    }
}

// ----------------------------------------------------------------- WMMA GEMM
// C[M, NTOT] = act(A[M, KTOT]) @ W[NTOT, KTOT]^T + bias
// act = identity, or (if BN_IN) relu(A*scale[k] + shift[k]).
// Each 256-thread block owns ROWS=32 rows and the FULL N range. The whole
// 32 x KTOT A-tile lands in LDS before any global write -> safe when C
// aliases A (in-place fine GEMM2).
// Wave layout: 8 waves along N (NTOT/8 cols each), each wave owns BOTH
// 16-row M-tiles -> every B fragment feeds 2 WMMAs (halved LDS traffic).
// W K-slices are register-double-buffered: slice kb+1 streams from global
// into VGPRs while slice kb is consumed by the matrix pipe.
// A staging: async DMA (GLOBAL_LOAD_ASYNC_TO_LDS_B128) when no per-element
// transform is needed; VALU path when BN+ReLU must be fused on load.
template <int KTOT, int NTOT, bool BN_IN>
__global__ __launch_bounds__(256) void k_gemm(const float* __restrict__ A,
                                              const float* __restrict__ W,
                                              const float* __restrict__ bias,
                                              float* __restrict__ C) {
    static_assert(KTOT % 16 == 0 && NTOT % 128 == 0, "tile config");
    constexpr int ROWS  = 32;
    constexpr int ASTR  = KTOT + 4;    // even stride; row pitch is 16B-aligned
    constexpr int BSTR  = 20;          // 16 K-slice + pad; 80B rows keep b128 stores aligned
    constexpr int NT    = NTOT / 128;  // 16x16 N-tiles per wave (wave covers NTOT/8 cols)
    constexpr int NREG4 = NTOT / 64;   // float4s per thread for one W K-slice
    constexpr int KB    = KTOT / 16;

    __shared__ float sA[ROWS * ASTR];
    __shared__ float sB[NTOT * BSTR];

    const int  tid = threadIdx.x;
    const long m0  = (long)blockIdx.x * ROWS;

    if constexpr (!BN_IN && ASYNC_LDS) {
        // Async DMA the A tile straight into LDS (no VGPR round-trip).
        constexpr int NCPY = ROWS * KTOT / 4;  // b128 transfers
        for (int i = tid; i < NCPY; i += 256) {
            int r = i / (KTOT / 4), q = i - r * (KTOT / 4);
            async_copy_b128(&A[(m0 + r) * KTOT + q * 4], &sA[r * ASTR + q * 4]);
        }
        async_wait0();  // our copies done; first __syncthreads publishes all waves'
    } else {
        // VALU path: fuse BN scale/shift + ReLU while staging (bases may be 4B-aligned).
        for (int i = tid; i < ROWS * KTOT; i += 256) {
            int r = i / KTOT, c = i - r * KTOT;
            float v = A[(m0 + r) * KTOT + c];
            if (BN_IN) {
                v = fmaf(v, g_scale[c], g_shift[c]);
                v = v > 0.0f ? v : 0.0f;
            }
            sA[r * ASTR + c] = v;
        }
    }

    const int lane = tid & 31;
    const int wvN  = tid >> 5;     // 8 waves along N
    const int half = lane >> 4;    // lane-half selects K pair (ISA A/B layout)
    const int l16  = lane & 15;

    v8f acc[2][NT];
#pragma unroll
    for (int mt = 0; mt < 2; ++mt)
#pragma unroll
        for (int t = 0; t < NT; ++t)
#pragma unroll
            for (int i = 0; i < 8; ++i) acc[mt][t][i] = 0.0f;

    // Prefetch W K-slice 0 into registers (vectorized; W is 256B-aligned input).
    v4f wreg[NREG4];
#pragma unroll
    for (int j = 0; j < NREG4; ++j) {
        int i = tid + j * 256;
        int n = i >> 2, q = i & 3;
        wreg[j] = *(const v4f*)&W[(size_t)n * KTOT + q * 4];
    }

    const float* sA0 = &sA[l16 * ASTR];
    const float* sA1 = &sA[(16 + l16) * ASTR];

    for (int kb = 0; kb < KB; ++kb) {
        __syncthreads();  // previous slice fully consumed (iter 0: sA stores/DMA visible)
#pragma unroll
        for (int j = 0; j < NREG4; ++j) {
            int i = tid + j * 256;
            int n = i >> 2, q = i & 3;
            *(v4f*)&sB[n * BSTR + q * 4] = wreg[j];
        }
        __syncthreads();

        // Stream slice kb+1 from global while the matrix pipe works on kb.
        if (kb + 1 < KB) {
#pragma unroll
            for (int j = 0; j < NREG4; ++j) {
                int i = tid + j * 256;
                int n = i >> 2, q = i & 3;
                wreg[j] = *(const v4f*)&W[(size_t)n * KTOT + (kb + 1) * 16 + q * 4];
            }
        }

#pragma unroll
        for (int km = 0; km < 4; ++km) {
            const int kcol = km * 4 + 2 * half;  // even -> b64 LDS load
            v2f a0 = *(const v2f*)&sA0[kb * 16 + kcol];
            v2f a1 = *(const v2f*)&sA1[kb * 16 + kcol];
#pragma unroll
            for (int t = 0; t < NT; ++t) {
                const float* sBrow = &sB[(wvN * (NTOT / 8) + t * 16 + l16) * BSTR];
                v2f b = *(const v2f*)&sBrow[kcol];
                // V_WMMA_F32_16X16X4_F32: D = A(16x4) x B(4x16) + C, full f32
                acc[0][t] = __builtin_amdgcn_wmma_f32_16x16x4_f32(
                    false, a0, false, b, (short)0, acc[0][t], false, false);
                acc[1][t] = __builtin_amdgcn_wmma_f32_16x16x4_f32(
                    false, a1, false, b, (short)0, acc[1][t], false, false);
            }
        }
    }

    // Epilogue: C/D layout -> VGPR r, lane 16h+n = element [M=r+8h][N=n]
#pragma unroll
    for (int mt = 0; mt < 2; ++mt) {
#pragma unroll
        for (int t = 0; t < NT; ++t) {
            const int col = wvN * (NTOT / 8) + t * 16 + l16;
            const float bv = bias[col];
#pragma unroll
            for (int r = 0; r < 8; ++r) {
                const long row = m0 + mt * 16 + r + 8 * half;
                C[row * NTOT + col] = acc[mt][t][r] + bv;
            }
        }
    }
}

// ------------------------------------------------------------ L2 row-norm
__global__ void k_l2norm(float* __restrict__ g, int D) {
    __shared__ float red[256];
    float* row = g + (size_t)blockIdx.x * D;
    float s = 0.0f;
    for (int c = threadIdx.x; c < D; c += 256) { float v = row[c]; s += v * v; }
    red[threadIdx.x] = s;
    __syncthreads();
    for (int off = 128; off > 0; off >>= 1) {
        if (threadIdx.x < off) red[threadIdx.x] += red[threadIdx.x + off];
        __syncthreads();
    }
    const float inv = rsqrtf(red[0]);
    for (int c = threadIdx.x; c < D; c += 256) row[c] *= inv;
}

// ---------------------------------------------------------------- top-k (K=5)
__global__ void k_topk(const float* __restrict__ attn) {
    int b = blockIdx.x * blockDim.x + threadIdx.x;
    if (b >= 256) return;
    const float* a = attn + b * 196;
    int chosen[5];
    for (int k = 0; k < 5; ++k) {
        float best = -1.0e30f; int bi = 0;
        for (int j = 0; j < 196; ++j) {
            bool skip = false;
            for (int p = 0; p < k; ++p) skip |= (chosen[p] == j);
            float v = a[j];
            if (!skip && v > best) { best = v; bi = j; }  // ties -> lowest index, like lax.top_k
        }
        chosen[k] = bi;
    }
    g_topidx[b * 6 + 0] = 0;                               // CLS
    for (int k = 0; k < 5; ++k) g_topidx[b * 6 + 1 + k] = chosen[k] + 1;
}

// -------------------------------------------------- gather top rows + avg row
__global__ void k_gather_avg(const float* __restrict__ Y, float* __restrict__ top) {
    const int b = blockIdx.x;
    const float* yb = Y + (size_t)b * 197 * 512;
    float* tb = top + (size_t)b * 7 * 512;
    for (int c = threadIdx.x; c < 512; c += blockDim.x) {
        float s = 0.0f;
        for (int n = 0; n < 197; ++n) s += yb[n * 512 + c];
        tb[6 * 512 + c] = s * (1.0f / 197.0f);
        for (int k = 0; k < 6; ++k) {
            const int idx = g_topidx[b * 6 + k];
            tb[k * 512 + c] = yb[idx * 512 + c];
        }
    }
}

// ------------------------------------------------------------------ launcher
extern "C" void kernel_launch(void* const* d_in, const int* in_sizes, int n_in,
                              void* d_out, int out_size, void* d_ws, size_t ws_size,
                              hipStream_t stream) {
    (void)in_sizes; (void)n_in; (void)out_size; (void)d_ws; (void)ws_size;

    const float* image_features = (const float*)d_in[0];   // [256,512]
    const float* fine_feat      = (const float*)d_in[1];   // [256,197,512]
    const float* fine_attn      = (const float*)d_in[2];   // [256,196]
    const float* logit_scale    = (const float*)d_in[3];   // scalar
    const float* vpt_w1  = (const float*)d_in[4];          // [768,512]
    const float* vpt_b1  = (const float*)d_in[5];
    const float* vpt_g1  = (const float*)d_in[6];
    const float* vpt_be1 = (const float*)d_in[7];
    const float* vpt_w2  = (const float*)d_in[8];          // [768,768]
    const float* vpt_b2  = (const float*)d_in[9];
    const float* mid_w1  = (const float*)d_in[10];         // [512,512]
    const float* mid_b1  = (const float*)d_in[11];
    const float* mid_g1  = (const float*)d_in[12];
    const float* mid_be1 = (const float*)d_in[13];
    const float* mid_w2  = (const float*)d_in[14];         // [512,512]
    const float* mid_b2  = (const float*)d_in[15];

    float* out   = (float*)d_out;
    float* gfeat = out;                         // [256,768]
    float* scal  = out + 196608;                // [1]
    float* fineY = out + 196609;                // [50432,512]; doubles as H scratch
    float* top   = out + 196609 + 25821184;     // [1792,512]; doubles as h1 scratch

    k_scalar_exp<<<1, 32, 0, stream>>>(logit_scale, scal);

    // ---- global (vpt) path: 256 x 512 -> 768 -> BN/ReLU -> 768 -> L2 norm
    k_gemm<512, 768, false><<<256 / 32, 256, 0, stream>>>(image_features, vpt_w1, vpt_b1, top);
    k_stats_init<<<3, 256, 0, stream>>>(768);
    k_stats_accum<<<1, 256, 0, stream>>>(top, 256, 768, 256);
    k_stats_final<<<3, 256, 0, stream>>>(vpt_g1, vpt_be1, 768, 1.0f / 256.0f);
    k_gemm<768, 768, true><<<256 / 32, 256, 0, stream>>>(top, vpt_w2, vpt_b2, gfeat);
    k_l2norm<<<256, 256, 0, stream>>>(gfeat, 768);

    // ---- fine path: 50432 x 512 -> 512 -> BN/ReLU -> 512 (in-place GEMM2)
    k_gemm<512, 512, false><<<50432 / 32, 256, 0, stream>>>(fine_feat, mid_w1, mid_b1, fineY);
    k_stats_init<<<2, 256, 0, stream>>>(512);
    k_stats_accum<<<(50432 + 511) / 512, 256, 0, stream>>>(fineY, 50432, 512, 512);
    k_stats_final<<<2, 256, 0, stream>>>(mid_g1, mid_be1, 512, 1.0f / 50432.0f);
    k_gemm<512, 512, true><<<50432 / 32, 256, 0, stream>>>(fineY, mid_w2, mid_b2, fineY);

    // ---- top-k gather + token average (after global path freed `top` scratch)
    k_topk<<<1, 256, 0, stream>>>(fine_attn);
    k_gather_avg<<<256, 256, 0, stream>>>(fineY, top);
}